// T3A_62105227100466
// MI455X (gfx1250) — compile-verified
//
#include <hip/hip_runtime.h>
#include <hip/hip_bf16.h>
#include <stdint.h>

typedef __attribute__((ext_vector_type(2))) float v2f;
typedef __attribute__((ext_vector_type(8))) float v8f;

#define WSZ 32
#define TPB 256
#define NWAVES (TPB / WSZ)
#define LIST_CAP 4096

// Monotone unsigned key for float ordering (handles negatives/NaN tails too).
__device__ __forceinline__ unsigned fkey(float f) {
  unsigned u = __float_as_uint(f);
  return (u & 0x80000000u) ? ~u : (u | 0x80000000u);
}

// ---------------- Phase 0: init scratch ----------------
__global__ void k_init(int* __restrict__ y_hat, int* __restrict__ counts,
                       int* __restrict__ cursor, int N, int C) {
  int id = blockIdx.x * blockDim.x + threadIdx.x;
  int stride = gridDim.x * blockDim.x;
  for (int i = id; i < N; i += stride) y_hat[i] = 0;
  if (id < C) { counts[id] = 0; cursor[id] = 0; }
}

// ---------------- Phase 1: one-hot argmax + class histogram ----------------
// labels is one-hot: argmax == position of the 1.0. Fully coalesced float4 scan.
__global__ void k_classify4(const float4* __restrict__ l4, int* __restrict__ y_hat,
                            int* __restrict__ counts, long long total4, int C) {
  long long id = (long long)blockIdx.x * blockDim.x + threadIdx.x;
  long long stride = (long long)gridDim.x * blockDim.x;
  for (long long t = id; t < total4; t += stride) {
    if (t + stride < total4) __builtin_prefetch(&l4[t + stride], 0, 0);
    float4 v = l4[t];
    long long base = t * 4;               // C % 4 == 0 -> all 4 in same row
    int row = (int)(base / C);
    int col = (int)(base - (long long)row * C);
    if (v.x > 0.5f) { y_hat[row] = col;     atomicAdd(&counts[col],     1); }
    if (v.y > 0.5f) { y_hat[row] = col + 1; atomicAdd(&counts[col + 1], 1); }
    if (v.z > 0.5f) { y_hat[row] = col + 2; atomicAdd(&counts[col + 2], 1); }
    if (v.w > 0.5f) { y_hat[row] = col + 3; atomicAdd(&counts[col + 3], 1); }
  }
}

__global__ void k_classify1(const float* __restrict__ labels, int* __restrict__ y_hat,
                            int* __restrict__ counts, long long total, int C) {
  long long id = (long long)blockIdx.x * blockDim.x + threadIdx.x;
  long long stride = (long long)gridDim.x * blockDim.x;
  for (long long t = id; t < total; t += stride) {
    if (labels[t] > 0.5f) {
      int row = (int)(t / C);
      int col = (int)(t - (long long)row * C);
      y_hat[row] = col;
      atomicAdd(&counts[col], 1);
    }
  }
}

// ---------------- Phase 2: exclusive prefix over C counters ----------------
__global__ void k_prefix(const int* __restrict__ counts, int* __restrict__ offsets, int C) {
  if (blockIdx.x == 0 && threadIdx.x == 0) {
    int s = 0;
    for (int i = 0; i < C; ++i) { offsets[i] = s; s += counts[i]; }
    offsets[C] = s;
  }
}

// ---------------- Phase 3: counting-sort scatter into per-class buckets ----------------
__global__ void k_scatter(const int* __restrict__ y_hat, const float* __restrict__ ent,
                          const int* __restrict__ offsets, int* __restrict__ cursor,
                          float* __restrict__ b_ent, int* __restrict__ b_idx, int N, int C) {
  int id = blockIdx.x * blockDim.x + threadIdx.x;
  int stride = gridDim.x * blockDim.x;
  for (int i = id; i < N; i += stride) {
    int c = y_hat[i];
    if (c < 0 || c >= C) continue;          // defensive
    int p = atomicAdd(&cursor[c], 1);
    int s = offsets[c] + p;
    b_ent[s] = ent[i];
    b_idx[s] = i;
  }
}

// ---------------- Phase 4: per-class top-K select + normalized accumulate ----------------
// One block per class. Radix-threshold (order-independent) selection, then a
// bitonic index-sort in LDS so FP accumulation order is deterministic.
__global__ void k_select_accum(const float* __restrict__ supports,
                               const float* __restrict__ b_ent,
                               const int* __restrict__ b_idx,
                               const int* __restrict__ offsets,
                               const int* __restrict__ pK,
                               float* __restrict__ Wm, int C, int D) {
  extern __shared__ float smem[];               // [NWAVES*D accum][TPB reduce]
  float* accum = smem;
  int*   ired  = (int*)(smem + NWAVES * D);
  float* fred  = (float*)ired;
  __shared__ int s_list[LIST_CAP];
  __shared__ unsigned s_tieCnt;
  __shared__ int s_nSel;

  const int c = blockIdx.x;
  const int tid = threadIdx.x;
  const int lane = tid & (WSZ - 1);
  const int wv = tid / WSZ;

  const int start = offsets[c];
  const int cnt = offsets[c + 1] - start;
  int K = pK[0];
  int need = cnt < K ? cnt : K;
  if (need > LIST_CAP) need = LIST_CAP;         // capacity clamp (K=100 here)

  for (int d = tid; d < NWAVES * D; d += blockDim.x) accum[d] = 0.f;
  if (tid == 0) { s_tieCnt = 0u; s_nSel = 0; }
  __syncthreads();

  // -- binary search for K-th smallest entropy key --
  unsigned T = 0xFFFFFFFFu;
  int tieNeed = cnt + 1;                        // cnt <= need: take everything
  if (cnt > need) {
    unsigned lo = 0u, hi = 0xFFFFFFFFu;
    for (int it = 0; it < 32; ++it) {
      unsigned mid = lo + ((hi - lo) >> 1);
      int cl = 0;
      for (int j = tid; j < cnt; j += blockDim.x)
        cl += (fkey(b_ent[start + j]) <= mid) ? 1 : 0;
      ired[tid] = cl; __syncthreads();
      for (int s = blockDim.x >> 1; s > 0; s >>= 1) {
        if (tid < s) ired[tid] += ired[tid + s];
        __syncthreads();
      }
      int tot = ired[0]; __syncthreads();
      if (tot >= need) hi = mid; else lo = mid + 1u;
    }
    T = hi;
    int cl = 0;
    for (int j = tid; j < cnt; j += blockDim.x)
      cl += (fkey(b_ent[start + j]) < T) ? 1 : 0;
    ired[tid] = cl; __syncthreads();
    for (int s = blockDim.x >> 1; s > 0; s >>= 1) {
      if (tid < s) ired[tid] += ired[tid + s];
      __syncthreads();
    }
    tieNeed = need - ired[0]; __syncthreads();
  }

  // -- build selected-index list --
  for (int j = tid; j < cnt; j += blockDim.x) {
    unsigned key = fkey(b_ent[start + j]);
    bool sel = key < T;
    if (!sel && key == T) {
      unsigned slot = atomicAdd(&s_tieCnt, 1u);
      sel = slot < (unsigned)tieNeed;
    }
    if (sel) {
      int p = atomicAdd(&s_nSel, 1);
      if (p < LIST_CAP) s_list[p] = b_idx[start + j];
    }
  }
  __syncthreads();
  int nSel = s_nSel; if (nSel > need) nSel = need;

  // -- bitonic sort of indices (deterministic accumulation order) --
  int P = 1; while (P < nSel) P <<= 1;
  for (int i = tid + nSel; i < P; i += blockDim.x) s_list[i] = 0x7FFFFFFF;
  __syncthreads();
  for (int k = 2; k <= P; k <<= 1)
    for (int jj = k >> 1; jj > 0; jj >>= 1) {
      for (int i = tid; i < P; i += blockDim.x) {
        int l = i ^ jj;
        if (l > i) {
          int a = s_list[i], b = s_list[l];
          bool up = ((i & k) == 0);
          if ((a > b) == up) { s_list[i] = b; s_list[l] = a; }
        }
      }
      __syncthreads();
    }

  // -- accumulate L2-normalized support rows: one row per wave --
  const int nseg = D / WSZ;                     // D=512 -> 16 segments/lane
  float* ac = accum + wv * D;
  for (int t = wv; t < nSel; t += NWAVES) {
    int r = s_list[t];
    const float* row = supports + (size_t)r * D;
    float v[16];
    float ss = 0.f;
    for (int s = 0; s < nseg; ++s) { float x = row[lane + s * WSZ]; v[s] = x; ss += x * x; }
    for (int o = WSZ >> 1; o >= 1; o >>= 1) ss += __shfl_xor(ss, o);
    float inv = (ss > 1e-24f) ? rsqrtf(ss) : 0.f;
    for (int s = 0; s < nseg; ++s) ac[lane + s * WSZ] += v[s] * inv;
  }
  __syncthreads();

  // -- fold wave accumulators, column-normalize, write W[:, c] --
  float colsq = 0.f;
  const int per = (D + blockDim.x - 1) / blockDim.x;   // 2 for D=512
  float mine[8];
  for (int p = 0; p < per; ++p) {
    int d = tid + p * blockDim.x;
    float a = 0.f;
    if (d < D) for (int w2 = 0; w2 < NWAVES; ++w2) a += accum[w2 * D + d];
    mine[p] = a;
    colsq += a * a;
  }
  fred[tid] = colsq; __syncthreads();
  for (int s = blockDim.x >> 1; s > 0; s >>= 1) {
    if (tid < s) fred[tid] += fred[tid + s];
    __syncthreads();
  }
  float nrm = sqrtf(fred[0]);
  float scl = (nrm > 1e-12f) ? (1.f / nrm) : 0.f;
  for (int p = 0; p < per; ++p) {
    int d = tid + p * blockDim.x;
    if (d < D) Wm[(size_t)d * C + c] = mine[p] * scl;
  }
}

// ---------------- Phase 5: out = z @ W via V_WMMA_F32_16X16X4_F32 ----------------
// One wave per 16x16 output tile; K-loop steps of 4, branchless clamp+mask
// edge handling so EXEC stays all-ones and loads need no exec gymnastics.
// A 16x4 layout: lane%16 = M, K = kk + vgpr + 2*(lane/16)
// B 4x16 layout: lane%16 = N, K = kk + vgpr + 2*(lane/16)
// D 16x16 layout: vgpr r -> M = r + 8*(lane/16), N = lane%16
__global__ void k_gemm_wmma(const float* __restrict__ z, const float* __restrict__ Wm,
                            float* __restrict__ out, int B, int D, int C) {
  const int m0 = blockIdx.x * 16;
  const int n0 = blockIdx.y * 16;
  const int lane = threadIdx.x;
  const int lhalf = lane >> 4;
  const int llo = lane & 15;

  const int mA = m0 + llo;
  const int nB = n0 + llo;
  const float amask = (mA < B) ? 1.f : 0.f;       // zero OOB rows of A
  const float bmask = (nB < C) ? 1.f : 0.f;       // zero OOB cols of B
  const int mAc = (mA < B) ? mA : (B - 1);        // clamped, always-legal addresses
  const int nBc = (nB < C) ? nB : (C - 1);

  const float* zp = z + (size_t)mAc * D + (lhalf << 1);   // += 4 per k-step
  const float* wp = Wm + (size_t)(lhalf << 1) * C + nBc;  // += 4*C per k-step
  const size_t wstep = (size_t)4 * C;

  v8f acc = {0.f, 0.f, 0.f, 0.f, 0.f, 0.f, 0.f, 0.f};
#pragma unroll 4
  for (int kk = 0; kk < D; kk += 4) {
    const float2 az = *(const float2*)zp;
    v2f a, b;
    a.x = az.x * amask;
    a.y = az.y * amask;
    b.x = wp[0] * bmask;
    b.y = wp[C] * bmask;
    acc = __builtin_amdgcn_wmma_f32_16x16x4_f32(
        /*neg_a=*/false, a, /*neg_b=*/false, b,
        /*c_mod=*/(short)0, acc, /*reuse_a=*/false, /*reuse_b=*/false);
    zp += 4;
    wp += wstep;
  }

#pragma unroll
  for (int r = 0; r < 8; ++r) {
    int m = m0 + r + (lhalf << 3);
    int n = n0 + llo;
    if (m < B && n < C) out[(size_t)m * C + n] = acc[r];
  }
}

// ---------------- Host launcher ----------------
static inline int imin(long long a, long long b) { return (int)(a < b ? a : b); }

extern "C" void kernel_launch(void* const* d_in, const int* in_sizes, int n_in,
                              void* d_out, int out_size, void* d_ws, size_t ws_size,
                              hipStream_t stream) {
  const float* z        = (const float*)d_in[0];
  const float* supports = (const float*)d_in[1];
  const float* labels   = (const float*)d_in[2];
  const float* ent      = (const float*)d_in[3];
  const int*   pK       = (const int*)d_in[4];     // device scalar, read in-kernel

  const int N = in_sizes[3];
  const int D = in_sizes[1] / N;
  const int C = in_sizes[2] / N;
  const int B = out_size / C;

  // carve workspace
  char* w = (char*)d_ws;
  auto carve = [&](size_t bytes) { char* p = w; w += (bytes + 255) & ~(size_t)255; return p; };
  int*   y_hat   = (int*)  carve((size_t)N * 4);
  int*   counts  = (int*)  carve((size_t)C * 4);
  int*   offsets = (int*)  carve((size_t)(C + 1) * 4);
  int*   cursor  = (int*)  carve((size_t)C * 4);
  float* b_ent   = (float*)carve((size_t)N * 4);
  int*   b_idx   = (int*)  carve((size_t)N * 4);
  float* Wm      = (float*)carve((size_t)D * C * 4);
  (void)ws_size; (void)n_in;

  const int blocksN = imin(((long long)N + TPB - 1) / TPB, 4096);
  k_init<<<blocksN, TPB, 0, stream>>>(y_hat, counts, cursor, N, C);

  const long long total = (long long)N * C;
  if ((C & 3) == 0) {
    const long long t4 = total / 4;
    const int blk = imin((t4 + TPB - 1) / TPB, 8192);
    k_classify4<<<blk, TPB, 0, stream>>>((const float4*)labels, y_hat, counts, t4, C);
  } else {
    const int blk = imin((total + TPB - 1) / TPB, 8192);
    k_classify1<<<blk, TPB, 0, stream>>>(labels, y_hat, counts, total, C);
  }

  k_prefix<<<1, 32, 0, stream>>>(counts, offsets, C);

  k_scatter<<<blocksN, TPB, 0, stream>>>(y_hat, ent, offsets, cursor, b_ent, b_idx, N, C);

  const size_t shmem = (size_t)(NWAVES * D + TPB) * sizeof(float);
  k_select_accum<<<C, TPB, shmem, stream>>>(supports, b_ent, b_idx, offsets, pK, Wm, C, D);

  dim3 g((B + 15) / 16, (C + 15) / 16);
  k_gemm_wmma<<<g, 32, 0, stream>>>(z, Wm, (float*)d_out, B, D, C);
}